// ConcatAttention_78305843740676
// MI455X (gfx1250) — compile-verified
//
#include <hip/hip_runtime.h>

typedef __attribute__((ext_vector_type(2))) float v2f;
typedef __attribute__((ext_vector_type(8))) float v8f;
typedef __attribute__((ext_vector_type(4))) int   v4i;
typedef __attribute__((address_space(1))) v4i     gv4i;  // global int4
typedef __attribute__((address_space(3))) v4i     lv4i;  // LDS int4

#define B_ 16
#define S_ 4096
#define H_ 512
#define D_ 1024

#define AROW 1028   // padded LDS row stride (floats): bank stride 4 -> conflict-free frag reads

#if defined(__has_builtin)
#if __has_builtin(__builtin_amdgcn_global_load_async_to_lds_b128) && \
    __has_builtin(__builtin_amdgcn_s_wait_asynccnt)
#define HAVE_ASYNC 1
#endif
#endif
#ifndef HAVE_ASYNC
#define HAVE_ASYNC 0
#endif

// ---------------------------------------------------------------------------
// K0: dec_proj[b,h] = sum_d dec[b,d] * W_a[h, 1024+d]; also zero attention out.
// ---------------------------------------------------------------------------
__global__ __launch_bounds__(256) void k_decproj(const float* __restrict__ dec,
                                                 const float* __restrict__ Wa,
                                                 float* __restrict__ dec_proj,
                                                 float* __restrict__ att) {
  int idx = blockIdx.x * 256 + threadIdx.x;     // 0..8191
  int b = idx >> 9;
  int h = idx & 511;
  const float* dh = dec + b * D_;
  const float* w  = Wa + (size_t)h * 2048 + D_; // W_dec row
  float acc = 0.f;
  for (int d = 0; d < D_; d += 4) {
    float4 x = *(const float4*)(dh + d);
    float4 y = *(const float4*)(w + d);
    acc += x.x * y.x + x.y * y.y + x.z * y.z + x.w * y.w;
  }
  dec_proj[idx] = acc;
  att[idx] = 0.f;
  att[idx + 8192] = 0.f;
}

// ---------------------------------------------------------------------------
// K1: fused scores kernel. One WG = 64 encoder rows x full H=512.
// 512 threads = 16 waves; wave w covers h in [w*32, w*32+32) as 2 WMMA N-tiles.
// A-tile (64 x 1024 f32) staged in LDS via async global->LDS.
// C = A(16x4) x B(4x16) accumulated over K=D via V_WMMA_F32_16X16X4_F32:
//   A[m,k]: lane = (k>>1)*16 + m, vgpr = k&1   (read from LDS, conflict-free)
//   B[k,n]: lane = (k>>1)*16 + n, vgpr = k&1   (W_enc rows, L2-resident)
//   C: N = lane&15, M = vgpr + 8*(lane>>4)
// scores[m] = sum_h v[h]*tanh(C + dec_proj[b,h]) via shuffles + LDS atomics.
// ---------------------------------------------------------------------------
__global__ __launch_bounds__(512) void k_scores(const float* __restrict__ enc,
                                                const unsigned char* __restrict__ emask,
                                                const float* __restrict__ Wa,
                                                const float* __restrict__ va,
                                                const float* __restrict__ dec_proj,
                                                float* __restrict__ scores) {
  __shared__ float lds[64 * AROW];
  __shared__ float sc[64];
  const int tid  = threadIdx.x;
  const int b    = blockIdx.y;
  const int s0   = blockIdx.x * 64;
  const int lane = tid & 31;
  const int wave = tid >> 5;
  const int m    = lane & 15;   // A-row / B-col / C-col index
  const int kg   = lane >> 4;   // K-pair group: 0 -> K{0,1}, 1 -> K{2,3}

  // ---- stage 64x1024 f32 encoder tile into LDS (16384 float4, 32/thread) ----
  const float* src = enc + ((size_t)b * S_ + s0) * D_;
#if HAVE_ASYNC
  for (int j = 0; j < 32; ++j) {
    int i   = j * 512 + tid;
    int row = i >> 8;             // 256 float4 per row
    int c4  = (i & 255) * 4;
    __builtin_amdgcn_global_load_async_to_lds_b128(
        (gv4i*)(src + (size_t)row * D_ + c4),
        (lv4i*)&lds[row * AROW + c4],
        0, 0);
  }
  __builtin_amdgcn_s_wait_asynccnt(0);
#else
  for (int j = 0; j < 32; ++j) {
    int i   = j * 512 + tid;
    int row = i >> 8;
    int c4  = (i & 255) * 4;
    *(float4*)&lds[row * AROW + c4] = *(const float4*)(src + (size_t)row * D_ + c4);
  }
#endif
  if (tid < 64) sc[tid] = 0.f;
  __syncthreads();

  // A fragments: rows (s0 + 16*mi + m), K-pair 2*kg, from LDS
  const float* ap0 = &lds[(m +  0) * AROW + 2 * kg];
  const float* ap1 = &lds[(m + 16) * AROW + 2 * kg];
  const float* ap2 = &lds[(m + 32) * AROW + 2 * kg];
  const float* ap3 = &lds[(m + 48) * AROW + 2 * kg];
  // B fragments: W_enc rows h = wave*32 + t*16 + m
  const size_t hb = (size_t)(wave * 32 + m);
  const float* wr0 = Wa + (hb +  0) * 2048 + 2 * kg;
  const float* wr1 = Wa + (hb + 16) * 2048 + 2 * kg;

  v8f c00 = {}, c01 = {}, c10 = {}, c11 = {};
  v8f c20 = {}, c21 = {}, c30 = {}, c31 = {};
  for (int k = 0; k < D_; k += 4) {
    v2f a0  = *(const v2f*)(ap0 + k);
    v2f a1  = *(const v2f*)(ap1 + k);
    v2f a2  = *(const v2f*)(ap2 + k);
    v2f a3  = *(const v2f*)(ap3 + k);
    v2f bv0 = *(const v2f*)(wr0 + k);
    v2f bv1 = *(const v2f*)(wr1 + k);
    c00 = __builtin_amdgcn_wmma_f32_16x16x4_f32(false, a0, false, bv0, (short)0, c00, false, false);
    c01 = __builtin_amdgcn_wmma_f32_16x16x4_f32(false, a0, false, bv1, (short)0, c01, false, false);
    c10 = __builtin_amdgcn_wmma_f32_16x16x4_f32(false, a1, false, bv0, (short)0, c10, false, false);
    c11 = __builtin_amdgcn_wmma_f32_16x16x4_f32(false, a1, false, bv1, (short)0, c11, false, false);
    c20 = __builtin_amdgcn_wmma_f32_16x16x4_f32(false, a2, false, bv0, (short)0, c20, false, false);
    c21 = __builtin_amdgcn_wmma_f32_16x16x4_f32(false, a2, false, bv1, (short)0, c21, false, false);
    c30 = __builtin_amdgcn_wmma_f32_16x16x4_f32(false, a3, false, bv0, (short)0, c30, false, false);
    c31 = __builtin_amdgcn_wmma_f32_16x16x4_f32(false, a3, false, bv1, (short)0, c31, false, false);
  }

  // scores[16*MI + r + 8*kg] += sum_n v[h]*tanh(C[r] + dec_proj[b,h])
#define REDUCE_TILE(CC, MI, TOFF)                                       \
  {                                                                     \
    int h = wave * 32 + (TOFF) + m;                                     \
    float dp = dec_proj[b * H_ + h];                                    \
    float vv = va[h];                                                   \
    _Pragma("unroll")                                                   \
    for (int r = 0; r < 8; ++r) {                                       \
      float x = vv * tanhf(CC[r] + dp);                                 \
      x += __shfl_xor(x, 1, 32);                                        \
      x += __shfl_xor(x, 2, 32);                                        \
      x += __shfl_xor(x, 4, 32);                                        \
      x += __shfl_xor(x, 8, 32);                                        \
      if (m == 0) atomicAdd(&sc[(MI) * 16 + r + kg * 8], x);            \
    }                                                                   \
  }
  REDUCE_TILE(c00, 0, 0)  REDUCE_TILE(c01, 0, 16)
  REDUCE_TILE(c10, 1, 0)  REDUCE_TILE(c11, 1, 16)
  REDUCE_TILE(c20, 2, 0)  REDUCE_TILE(c21, 2, 16)
  REDUCE_TILE(c30, 3, 0)  REDUCE_TILE(c31, 3, 16)
#undef REDUCE_TILE

  __syncthreads();
  if (tid < 64) {
    int s = s0 + tid;
    float v = emask[b * S_ + s] ? -__builtin_inff() : sc[tid];
    scores[b * S_ + s] = v;
  }
}

// ---------------------------------------------------------------------------
// K2: per-batch masked softmax over S. One block per b.
// ---------------------------------------------------------------------------
__global__ __launch_bounds__(256) void k_softmax(const float* __restrict__ scores,
                                                 float* __restrict__ dist) {
  __shared__ float red[256];
  const int b = blockIdx.x, tid = threadIdx.x;
  const float* sr = scores + b * S_;
  float* dr = dist + b * S_;

  float mx = -__builtin_inff();
  for (int s = tid; s < S_; s += 256) mx = fmaxf(mx, sr[s]);
  red[tid] = mx;
  __syncthreads();
  for (int o = 128; o > 0; o >>= 1) {
    if (tid < o) red[tid] = fmaxf(red[tid], red[tid + o]);
    __syncthreads();
  }
  mx = red[0];
  __syncthreads();

  float sum = 0.f;
  for (int s = tid; s < S_; s += 256) {
    float e = expf(sr[s] - mx);   // exp(-inf) = 0 handles masked slots
    dr[s] = e;
    sum += e;
  }
  red[tid] = sum;
  __syncthreads();
  for (int o = 128; o > 0; o >>= 1) {
    if (tid < o) red[tid] += red[tid + o];
    __syncthreads();
  }
  float inv = 1.f / red[0];
  for (int s = tid; s < S_; s += 256) dr[s] *= inv;
}

// ---------------------------------------------------------------------------
// K3: attention[b,d] += sum_{s in chunk} enc[b,s,d] * dist[b,s]
// grid (D/256, S/512, B); coalesced enc streaming, weights staged in LDS.
// ---------------------------------------------------------------------------
__global__ __launch_bounds__(256) void k_attn(const float* __restrict__ enc,
                                              const float* __restrict__ dist,
                                              float* __restrict__ att) {
  __shared__ float pw[512];
  const int b  = blockIdx.z;
  const int s0 = blockIdx.y * 512;
  const int d  = blockIdx.x * 256 + threadIdx.x;
  for (int i = threadIdx.x; i < 512; i += 256) pw[i] = dist[b * S_ + s0 + i];
  __syncthreads();

  const float* ep = enc + ((size_t)b * S_ + s0) * D_ + d;
  float acc = 0.f;
#pragma unroll 8
  for (int j = 0; j < 512; ++j) acc += ep[(size_t)j * D_] * pw[j];
  atomicAdd(&att[b * D_ + d], acc);
}

// ---------------------------------------------------------------------------
// K4: zero attention rows where decoder_mask[b].
// ---------------------------------------------------------------------------
__global__ __launch_bounds__(256) void k_mask(const unsigned char* __restrict__ dmask,
                                              float* __restrict__ att) {
  int idx = blockIdx.x * 256 + threadIdx.x;
  if (idx < B_ * D_ && dmask[idx >> 10]) att[idx] = 0.f;
}

// ---------------------------------------------------------------------------
extern "C" void kernel_launch(void* const* d_in, const int* in_sizes, int n_in,
                              void* d_out, int out_size, void* d_ws, size_t ws_size,
                              hipStream_t stream) {
  const float*         enc   = (const float*)d_in[0];
  const unsigned char* emask = (const unsigned char*)d_in[1]; // bool (1 byte)
  const float*         dec   = (const float*)d_in[2];
  const unsigned char* dmask = (const unsigned char*)d_in[3]; // bool (1 byte)
  const float*         Wa    = (const float*)d_in[4];
  const float*         va    = (const float*)d_in[5];

  float* att  = (float*)d_out;            // [B, D]
  float* dist = (float*)d_out + B_ * D_;  // [B, S]

  float* scores   = (float*)d_ws;             // B*S floats (256 KB)
  float* dec_proj = (float*)d_ws + B_ * S_;   // B*H floats (32 KB)

  k_decproj<<<dim3(32), dim3(256), 0, stream>>>(dec, Wa, dec_proj, att);

  k_scores<<<dim3(S_ / 64, B_), dim3(512), 0, stream>>>(enc, emask, Wa, va,
                                                        dec_proj, scores);

  k_softmax<<<dim3(B_), dim3(256), 0, stream>>>(scores, dist);

  k_attn<<<dim3(D_ / 256, S_ / 512, B_), dim3(256), 0, stream>>>(enc, dist, att);

  k_mask<<<dim3((B_ * D_ + 255) / 256), dim3(256), 0, stream>>>(dmask, att);
}